// PathSignatureComputer_48473000902841
// MI455X (gfx1250) — compile-verified
//
#include <hip/hip_runtime.h>

typedef __attribute__((ext_vector_type(2))) float v2f;
typedef __attribute__((ext_vector_type(8))) float v8f;

#define WIN 64
#define DPATH 32
#define LSTR 40                 // LDS row stride (floats): rows t, t+2 hit disjoint banks
#define OUT_PER_B (32 + 1024 + 32768)

__global__ __launch_bounds__(256)
void pathsig_wmma_kernel(const float* __restrict__ features,
                         const int* __restrict__ t_ptr,
                         float* __restrict__ out) {
  __shared__ float s_path[WIN * DPATH];
  __shared__ float s_inc [WIN * LSTR];
  __shared__ float s_prev[WIN * LSTR];

  const int b    = blockIdx.x;
  const int tid  = threadIdx.x;
  const int lane = tid & 31;
  const int wave = tid >> 5;

  const int t     = t_ptr[0];
  const int start = max(0, t - (WIN - 1));
  const int cur   = min(WIN, t + 1);
  const int pad   = WIN - cur;   // replicated leading rows when t < 63

  // ---- stage path (time channel + 31 features) into LDS ----
  for (int idx = tid; idx < WIN * DPATH; idx += 256) {
    const int s = idx >> 5;
    const int c = idx & 31;
    float v;
    if (c == 0) {
      v = (float)s * (1.0f / 63.0f);
    } else {
      const int src = start + max(0, s - pad);
      v = features[((long)b * 1024 + src) * 31 + (c - 1)];
    }
    s_path[idx] = v;
  }
  __syncthreads();

  // ---- level 1: path[63] - path[0] ----
  if (tid < 32) {
    out[(long)b * OUT_PER_B + tid] = s_path[63 * DPATH + tid] - s_path[tid];
  }

  // ---- inc (K-padded: row 63 = 0) and prev, bank-conflict-free stride ----
  for (int idx = tid; idx < WIN * DPATH; idx += 256) {
    const int s = idx >> 5;
    const int c = idx & 31;
    const float p  = s_path[idx];
    const float nx = (s < WIN - 1) ? s_path[idx + DPATH] : p;
    s_inc [s * LSTR + c] = nx - p;   // row 63 -> 0, zero-pads K to 64
    s_prev[s * LSTR + c] = p;
  }
  __syncthreads();

  const int m    = lane & 15;          // row/col within 16-wide tile
  const int koff = (lane >> 4) << 1;   // lanes 16-31 hold K+2, K+3

  // ---- sig2 = inc^T(32x64) @ prev(64x32), waves 0..3, one 16x16 tile each ----
  if (wave < 4) {
    const int mi = (wave >> 1) << 4;
    const int ni = (wave & 1) << 4;
    v8f c2 = {};
#pragma unroll 4
    for (int kk = 0; kk < 16; ++kk) {
      const int t0 = 4 * kk + koff;
      v2f a, bb;
      a.x  = s_inc [ t0      * LSTR + mi + m];
      a.y  = s_inc [(t0 + 1) * LSTR + mi + m];
      bb.x = s_prev[ t0      * LSTR + ni + m];
      bb.y = s_prev[(t0 + 1) * LSTR + ni + m];
      c2 = __builtin_amdgcn_wmma_f32_16x16x4_f32(false, a, false, bb,
                                                 (short)0, c2, false, false);
    }
    float* o2 = out + (long)b * OUT_PER_B + 32;
    const int mb = mi + ((lane >> 4) << 3);
#pragma unroll
    for (int r = 0; r < 8; ++r) {
      o2[(mb + r) * 32 + ni + m] = c2[r];
    }
  }
  // LDS is read-only from here on: no barrier needed before sig3.

  // ---- sig3 = G^T(1024x64) @ prev(64x32), G[t,ij] = inc[t,i]*prev[t,j] ----
  // B operand is invariant across M-tiles: hoist all 16 k-steps x 2 N-tiles
  // into registers once per wave, pre-scaled by 1/63 (sig3-only scale).
  const float sc = 1.0f / 63.0f;
  v2f b0r[16], b1r[16];
#pragma unroll
  for (int kk = 0; kk < 16; ++kk) {
    const int t0 = 4 * kk + koff;
    b0r[kk].x = s_prev[ t0      * LSTR + m]      * sc;
    b0r[kk].y = s_prev[(t0 + 1) * LSTR + m]      * sc;
    b1r[kk].x = s_prev[ t0      * LSTR + 16 + m] * sc;
    b1r[kk].y = s_prev[(t0 + 1) * LSTR + 16 + m] * sc;
  }

  float* o3 = out + (long)b * OUT_PER_B + 32 + 1024;
  for (int m0 = wave * 8; m0 < wave * 8 + 8; ++m0) {
    const int row = m0 * 16 + m;     // ij pair for this lane's A rows
    const int i   = row >> 5;        // constant within the 16-row tile
    const int j   = row & 31;
    v8f c0 = {}, c1 = {};
#pragma unroll
    for (int kk = 0; kk < 16; ++kk) {
      const int t0 = 4 * kk + koff;
      const int t1 = t0 + 1;
      v2f a;
      a.x = s_inc[t0 * LSTR + i] * s_prev[t0 * LSTR + j];
      a.y = s_inc[t1 * LSTR + i] * s_prev[t1 * LSTR + j];
      c0 = __builtin_amdgcn_wmma_f32_16x16x4_f32(false, a, false, b0r[kk],
                                                 (short)0, c0, false, false);
      c1 = __builtin_amdgcn_wmma_f32_16x16x4_f32(false, a, false, b1r[kk],
                                                 (short)0, c1, false, false);
    }
    const int mb = m0 * 16 + ((lane >> 4) << 3);
#pragma unroll
    for (int r = 0; r < 8; ++r) {
      o3[(mb + r) * 32 + m]      = c0[r];
      o3[(mb + r) * 32 + 16 + m] = c1[r];
    }
  }
}

extern "C" void kernel_launch(void* const* d_in, const int* in_sizes, int n_in,
                              void* d_out, int out_size, void* d_ws, size_t ws_size,
                              hipStream_t stream) {
  const float* features = (const float*)d_in[0];
  const int*   t_ptr    = (const int*)d_in[1];
  float*       out      = (float*)d_out;
  const int B = in_sizes[0] / (1024 * 31);   // 2048
  pathsig_wmma_kernel<<<B, 256, 0, stream>>>(features, t_ptr, out);
}